// Model_pro_65352222376313
// MI455X (gfx1250) — compile-verified
//
#include <hip/hip_runtime.h>

typedef float v2f __attribute__((ext_vector_type(2)));
typedef float v8f __attribute__((ext_vector_type(8)));

#define NG 48
#define GRID_STEP 0.5f

__global__ void zero_out_kernel(float* __restrict__ out, int n) {
    int i = blockIdx.x * blockDim.x + threadIdx.x;
    if (i < n) out[i] = 0.0f;
}

// One wave (32 lanes) per atom. WMMA f32 16x16x4 computes the outer sum
// dx2[i] + dy2[j] over the 16x16 xy-tile in a single matrix op; the D-matrix
// VGPR layout directly assigns 8 (i,j) cells per lane. A uniform z-loop adds
// dz2 and scatter-adds via hardware global_atomic_add_f32 (grid is L2-resident).
//
// Templated on species so every radius-derived coefficient is a compile-time
// literal (no runtime v_div_scale/v_rcp sequences in the wave preamble).
template <int SPEC>
__global__ __launch_bounds__(256) void splat_wmma_kernel(
    const float* __restrict__ vecs,
    float* __restrict__ outs,   // already offset to this species' 48^3 block
    int nAtoms)
{
    constexpr double Rd  = (SPEC == 0) ? 1.7 : ((SPEC == 1) ? 1.55 : 1.52);
    constexpr double E2d = 7.38905609893065;  // exp(2)
    constexpr float  R    = (float)Rd;
    constexpr float  BB   = (float)(1.5 * Rd);
    constexpr float  R2   = (float)(Rd * Rd);
    constexpr float  B2   = (float)(1.5 * Rd * 1.5 * Rd);
    constexpr float  EXPK = (float)(-2.0 / (Rd * Rd));
    constexpr float  C2   = (float)(4.0 / (E2d * Rd * Rd));
    constexpr float  C1   = (float)(12.0 / (E2d * Rd));
    constexpr float  C0   = (float)(9.0 / E2d);
    (void)R;

    const int lane = threadIdx.x & 31;
    const int wave = (blockIdx.x << 3) + (threadIdx.x >> 5);  // 8 waves/block

    const bool live = (wave < nAtoms);
    const int  aidx = live ? wave : 0;

    // shifted coords: vec + n_grid*0.5 - 0.5
    const float vx = vecs[3 * aidx + 0] + (NG * 0.5f - 0.5f);
    const float vy = vecs[3 * aidx + 1] + (NG * 0.5f - 0.5f);
    const float vz = vecs[3 * aidx + 2] + (NG * 0.5f - 0.5f);

    // int() truncates toward zero (matches astype(int32)); clip to [0, NG]
    int minx = min(max((int)((vx - BB) * 2.0f), 0), NG);
    int miny = min(max((int)((vy - BB) * 2.0f), 0), NG);
    int minz = min(max((int)((vz - BB) * 2.0f), 0), NG);
    int maxx = min(max((int)(2.0f + (vx + BB) * 2.0f), 0), NG);
    int maxy = min(max((int)(2.0f + (vy + BB) * 2.0f), 0), NG);
    int maxz = min(max((int)(2.0f + (vz + BB) * 2.0f), 0), NG);

    // ---- Build WMMA fragments for the outer-sum matmul ----
    // A (16x4): row M=lane (lanes 0-15), VGPR0=K0 -> dx2[M], VGPR1=K1 -> 1.0
    //           lanes 16-31 carry K2/K3 = 0.
    // B (4x16): VGPR0 lanes0-15 = row K0 = 1.0 ; VGPR1 lanes0-15 = row K1 = dy2[N]
    //           lanes 16-31 carry rows K2/K3 = 0.
    const bool lo = (lane < 16);
    const float dxv = vx - GRID_STEP * (float)(minx + lane);
    const float dyv = vy - GRID_STEP * (float)(miny + lane);

    v2f A, B;
    A.x = lo ? dxv * dxv : 0.0f;
    A.y = lo ? 1.0f      : 0.0f;
    B.x = lo ? 1.0f      : 0.0f;
    B.y = lo ? dyv * dyv : 0.0f;

    v8f acc = {};
    // D[i][j] = dx2[i]*1 + 1*dy2[j]  (K=2,3 contribute zero)
    acc = __builtin_amdgcn_wmma_f32_16x16x4_f32(
        /*neg_a=*/false, A, /*neg_b=*/false, B,
        /*c_mod=*/(short)0, acc, /*reuse_a=*/false, /*reuse_b=*/false);

    // ---- Per-element (i,j) setup from the D-matrix layout ----
    const int jcol  = lane & 15;          // N
    const int rbase = (lane >> 4) << 3;   // M offset: lanes 16-31 -> +8

    int  basexy[8];
    bool mxy[8];
#pragma unroll
    for (int v = 0; v < 8; ++v) {
        const int gx = minx + v + rbase;  // M = v + rbase
        const int gy = miny + jcol;       // N = jcol
        mxy[v]    = (gx < maxx) & (gy < maxy);
        basexy[v] = (gx * NG + gy) * NG;
    }

    int nz = maxz - minz;
    if (!live) nz = 0;                    // dead wave: WMMA ran (EXEC all-1s), no writes

    for (int z = 0; z < nz; ++z) {        // wave-uniform trip count, no divergence
        const int   gz  = minz + z;
        const float dzv = vz - GRID_STEP * (float)gz;
        const float dz2 = dzv * dzv;
#pragma unroll
        for (int v = 0; v < 8; ++v) {
            const float d2 = acc[v] + dz2;
            if (mxy[v] && d2 < B2) {      // skip outside-sphere cells (val == 0)
                float val;
                if (d2 < R2) {
                    val = __expf(EXPK * d2);                       // f1
                } else {
                    const float d = __builtin_sqrtf(d2);
                    val = C2 * d2 - C1 * d + C0;                   // f2
                }
#if defined(__HIP_DEVICE_COMPILE__)
                unsafeAtomicAdd(&outs[basexy[v] + gz], val);       // global_atomic_add_f32
#else
                atomicAdd(&outs[basexy[v] + gz], val);
#endif
            }
        }
    }
}

extern "C" void kernel_launch(void* const* d_in, const int* in_sizes, int n_in,
                              void* d_out, int out_size, void* d_ws, size_t ws_size,
                              hipStream_t stream) {
    const float* vC = (const float*)d_in[0];
    const float* vN = (const float*)d_in[1];
    const float* vO = (const float*)d_in[2];
    float* out = (float*)d_out;

    const int nAtoms = in_sizes[0] / 3;   // 16384
    const int SPECSZ = NG * NG * NG;

    // Zero the output grid (harness poisons it before timing).
    {
        int threads = 256;
        int blocks  = (out_size + threads - 1) / threads;
        zero_out_kernel<<<blocks, threads, 0, stream>>>(out, out_size);
    }

    // One wave per atom; 8 waves per 256-thread block; one launch per species
    // so radius coefficients are compile-time constants.
    const int blocks = (nAtoms + 7) / 8;
    splat_wmma_kernel<0><<<blocks, 256, 0, stream>>>(vC, out + 0 * SPECSZ, nAtoms);
    splat_wmma_kernel<1><<<blocks, 256, 0, stream>>>(vN, out + 1 * SPECSZ, nAtoms);
    splat_wmma_kernel<2><<<blocks, 256, 0, stream>>>(vO, out + 2 * SPECSZ, nAtoms);
}